// VQVAE_5076651344355
// MI455X (gfx1250) — compile-verified
//
#include <hip/hip_runtime.h>
#include <hip/hip_bf16.h>

typedef __attribute__((ext_vector_type(2))) float v2f;
typedef __attribute__((ext_vector_type(8))) float v8f;

__device__ __forceinline__ float lrelu(float x) { return x > 0.0f ? x : 0.2f * x; }

// ---------------- encoder ----------------

// conv1: 1 -> 32, k=4, s=2, p=1, T 8192 -> 4096, + lrelu
__global__ __launch_bounds__(256) void enc_conv1(const float* __restrict__ x,
                                                 const float* __restrict__ w,
                                                 const float* __restrict__ bias,
                                                 float* __restrict__ out) {
    int i = blockIdx.x * 256 + threadIdx.x;           // (b, oc, t), t fastest
    int t  = i & 4095;
    int oc = (i >> 12) & 31;
    int b  = i >> 17;
    const float* xb = x + (size_t)b * 8192;
    float acc = bias[oc];
    int s0 = 2 * t - 1;
#pragma unroll
    for (int k = 0; k < 4; ++k) {
        int s = s0 + k;
        float xv = (s >= 0 && s < 8192) ? xb[s] : 0.0f;
        acc += w[oc * 4 + k] * xv;
    }
    out[i] = lrelu(acc);
}

// generic 32->32, k=4, s=2, p=1, + lrelu
__global__ __launch_bounds__(256) void enc_conv_s2(const float* __restrict__ in,
                                                   const float* __restrict__ w,
                                                   const float* __restrict__ bias,
                                                   float* __restrict__ out,
                                                   int Tin, int Tout) {
    int i = blockIdx.x * 256 + threadIdx.x;
    int t  = i % Tout;
    int oc = (i / Tout) & 31;
    int b  = i / (Tout * 32);
    const float* inb = in + (size_t)b * 32 * Tin;
    float acc = bias[oc];
    int s0 = 2 * t - 1;
    for (int ic = 0; ic < 32; ++ic) {
        const float* r  = inb + (size_t)ic * Tin;
        const float* wr = w + (oc * 32 + ic) * 4;
#pragma unroll
        for (int k = 0; k < 4; ++k) {
            int s = s0 + k;
            float xv = (s >= 0 && s < Tin) ? r[s] : 0.0f;
            acc += wr[k] * xv;
        }
    }
    out[i] = lrelu(acc);
}

// conv4: 32 -> 64, k=3, s=1, p=1, T=1024, no activation.
// Writes z as [row=b*1024+t][d=oc] row-major (WMMA-A friendly).
__global__ __launch_bounds__(256) void enc_conv4(const float* __restrict__ in,
                                                 const float* __restrict__ w,
                                                 const float* __restrict__ bias,
                                                 float* __restrict__ z) {
    int i  = blockIdx.x * 256 + threadIdx.x;          // (b, t, oc), oc fastest
    int oc = i & 63;
    int t  = (i >> 6) & 1023;
    int b  = i >> 16;
    const float* inb = in + (size_t)b * 32 * 1024;
    float acc = bias[oc];
    for (int ic = 0; ic < 32; ++ic) {
        const float* r  = inb + (size_t)ic * 1024;
        const float* wr = w + (oc * 32 + ic) * 3;
#pragma unroll
        for (int k = 0; k < 3; ++k) {
            int s = t + k - 1;
            float xv = (s >= 0 && s < 1024) ? r[s] : 0.0f;
            acc += wr[k] * xv;
        }
    }
    z[i] = acc;                                        // i == row*64 + oc
}

// ---------------- vector quantization ----------------

// embedT[n][d] = embed[d][n]  (one-time 512 KB transpose; coalesced reads)
__global__ __launch_bounds__(256) void embed_transpose(const float* __restrict__ embed,
                                                       float* __restrict__ embedT) {
    int i = blockIdx.x * 256 + threadIdx.x;           // 64 * 2048
    int n = i & 2047;
    int d = i >> 11;
    embedT[(size_t)n * 64 + d] = embed[i];
}

// e2[n] = sum_d embed[d][n]^2  (coalesced over n)
__global__ __launch_bounds__(256) void embed_norm(const float* __restrict__ embed,
                                                  float* __restrict__ e2) {
    int n = blockIdx.x * 256 + threadIdx.x;
    if (n >= 2048) return;
    float s = 0.0f;
    for (int d = 0; d < 64; ++d) {
        float v = embed[(size_t)d * 2048 + n];
        s += v * v;
    }
    e2[n] = s;
}

// One wave per 16-row stripe of z (65536 x 64). Scores via V_WMMA_F32_16X16X4_F32
// against the codebook (embedT, n-major); running argmin of e2[n] - 2*score.
// B fragment per lane per k-step is 2 contiguous floats -> b64 loads.
__global__ __launch_bounds__(256) void vq_argmin(const float* __restrict__ z,
                                                 const float* __restrict__ embedT,
                                                 const float* __restrict__ e2,
                                                 int* __restrict__ idx_out) {
    const int lane   = threadIdx.x & 31;
    const int wave   = threadIdx.x >> 5;
    const int stripe = blockIdx.x * 8 + wave;
    const int row0   = stripe * 16;
    const int m      = lane & 15;
    const int half   = lane >> 4;

    // Preload A stripe: lane holds row m; VGPR v of k-step kk is K = kk*4 + 2*half + v.
    v2f a[16];
    const float* zr = z + (size_t)(row0 + m) * 64;
#pragma unroll
    for (int kk = 0; kk < 16; ++kk) {
        int k0 = kk * 4 + half * 2;
        a[kk] = *(const v2f*)(zr + k0);                // 8B contiguous pair
    }

    float best[8];
    int   bidx[8];
#pragma unroll
    for (int r = 0; r < 8; ++r) { best[r] = 3.4e38f; bidx[r] = 0; }

    for (int nt = 0; nt < 128; ++nt) {
        const int n = nt * 16 + m;
        const float* ecol = embedT + (size_t)n * 64;   // codebook row n, d-contiguous
        v8f c = {0.f, 0.f, 0.f, 0.f, 0.f, 0.f, 0.f, 0.f};
#pragma unroll
        for (int kk = 0; kk < 16; ++kk) {
            int k0 = kk * 4 + half * 2;
            v2f bb = *(const v2f*)(ecol + k0);         // 8B contiguous pair
            c = __builtin_amdgcn_wmma_f32_16x16x4_f32(false, a[kk], false, bb,
                                                      (short)0, c, false, false);
        }
        float en = e2[n];
#pragma unroll
        for (int r = 0; r < 8; ++r) {
            float d = en - 2.0f * c[r];                // row = r + 8*half, col = n
            if (d < best[r]) { best[r] = d; bidx[r] = n; }
        }
    }

    // min+index reduction within each 16-lane half (xor < 16 never crosses halves)
#pragma unroll
    for (int r = 0; r < 8; ++r) {
        float v = best[r];
        int   i = bidx[r];
#pragma unroll
        for (int offs = 1; offs < 16; offs <<= 1) {
            float ov = __shfl_xor(v, offs, 32);
            int   oi = __shfl_xor(i, offs, 32);
            if (ov < v || (ov == v && oi < i)) { v = ov; i = oi; }
        }
        if (m == 0) idx_out[row0 + r + half * 8] = i;
    }
}

// Gather quant rows into q[b][d][t], accumulate diff = mean((quant - z)^2).
// Gather from embedT is 64 contiguous floats per codebook row (coalesced).
__global__ __launch_bounds__(256) void vq_quant(const float* __restrict__ z,
                                                const float* __restrict__ embedT,
                                                const int* __restrict__ idx,
                                                float* __restrict__ q,
                                                float* __restrict__ diff_out) {
    int d    = threadIdx.x & 63;
    int rloc = threadIdx.x >> 6;
    int row  = blockIdx.x * 4 + rloc;
    int b = row >> 10, t = row & 1023;
    int id = idx[row];
    float qv = embedT[(size_t)id * 64 + d];
    float zv = z[(size_t)row * 64 + d];
    q[((size_t)(b * 64 + d)) * 1024 + t] = qv;
    float dd = qv - zv;
    dd = dd * dd;
    __shared__ float red[256];
    red[threadIdx.x] = dd;
    __syncthreads();
    for (int s = 128; s > 0; s >>= 1) {
        if ((int)threadIdx.x < s) red[threadIdx.x] += red[threadIdx.x + s];
        __syncthreads();
    }
    if (threadIdx.x == 0)
        atomicAdd(diff_out, red[0] * (1.0f / (65536.0f * 64.0f)));
}

__global__ void zero_scalar(float* p) {
    if (threadIdx.x == 0 && blockIdx.x == 0) *p = 0.0f;
}

// ---------------- decoder ----------------

// dw1: 64 -> 32, k=3, s=1, p=1, T=1024, + lrelu
__global__ __launch_bounds__(256) void dec_conv1(const float* __restrict__ q,
                                                 const float* __restrict__ w,
                                                 const float* __restrict__ bias,
                                                 float* __restrict__ out) {
    int i  = blockIdx.x * 256 + threadIdx.x;          // (b, oc, t)
    int t  = i & 1023;
    int oc = (i >> 10) & 31;
    int b  = i >> 15;
    const float* qb = q + (size_t)b * 64 * 1024;
    float acc = bias[oc];
    for (int ic = 0; ic < 64; ++ic) {
        const float* r  = qb + (size_t)ic * 1024;
        const float* wr = w + (oc * 64 + ic) * 3;
#pragma unroll
        for (int k = 0; k < 3; ++k) {
            int s = t + k - 1;
            float xv = (s >= 0 && s < 1024) ? r[s] : 0.0f;
            acc += wr[k] * xv;
        }
    }
    out[i] = lrelu(acc);
}

// ConvTranspose1d 32 -> 32, k=4, s=2, p=1, + lrelu. Weight layout (in, out, k).
__global__ __launch_bounds__(256) void dec_convT(const float* __restrict__ in,
                                                 const float* __restrict__ w,
                                                 const float* __restrict__ bias,
                                                 float* __restrict__ out,
                                                 int Tin) {
    int Tout = Tin * 2;
    int i  = blockIdx.x * 256 + threadIdx.x;
    int j  = i % Tout;
    int oc = (i / Tout) & 31;
    int b  = i / (Tout * 32);
    const float* inb = in + (size_t)b * 32 * Tin;
    // out[j] = taps with input index i_in = (j + 1 - k) / 2 when integral
    int ka, kb, ia, ib;
    if ((j & 1) == 0) { ka = 1; ia = j >> 1;       kb = 3; ib = (j >> 1) - 1; }
    else              { ka = 0; ia = (j + 1) >> 1; kb = 2; ib = (j - 1) >> 1; }
    bool va = (ia >= 0 && ia < Tin);
    bool vb = (ib >= 0 && ib < Tin);
    float acc = bias[oc];
    for (int ic = 0; ic < 32; ++ic) {
        const float* wr = w + (ic * 32 + oc) * 4;
        const float* r  = inb + (size_t)ic * Tin;
        if (va) acc += wr[ka] * r[ia];
        if (vb) acc += wr[kb] * r[ib];
    }
    out[i] = lrelu(acc);
}

// ConvTranspose1d 32 -> 1, k=4, s=2, p=1, no activation. Output (B, 8192).
__global__ __launch_bounds__(256) void dec_convT_last(const float* __restrict__ in,
                                                      const float* __restrict__ w,
                                                      const float* __restrict__ bias,
                                                      float* __restrict__ out,
                                                      int Tin) {
    int Tout = Tin * 2;                                // 8192
    int i = blockIdx.x * 256 + threadIdx.x;
    int j = i % Tout;
    int b = i / Tout;
    const float* inb = in + (size_t)b * 32 * Tin;
    int ka, kb, ia, ib;
    if ((j & 1) == 0) { ka = 1; ia = j >> 1;       kb = 3; ib = (j >> 1) - 1; }
    else              { ka = 0; ia = (j + 1) >> 1; kb = 2; ib = (j - 1) >> 1; }
    bool va = (ia >= 0 && ia < Tin);
    bool vb = (ib >= 0 && ib < Tin);
    float acc = bias[0];
    for (int ic = 0; ic < 32; ++ic) {
        const float* wr = w + ic * 4;                  // (ic, 0, k)
        const float* r  = inb + (size_t)ic * Tin;
        if (va) acc += wr[ka] * r[ia];
        if (vb) acc += wr[kb] * r[ib];
    }
    out[i] = acc;                                      // i == b*8192 + j
}

// ---------------- launch ----------------

extern "C" void kernel_launch(void* const* d_in, const int* in_sizes, int n_in,
                              void* d_out, int out_size, void* d_ws, size_t ws_size,
                              hipStream_t stream) {
    const float* x     = (const float*)d_in[0];
    const float* ew1   = (const float*)d_in[1];
    const float* eb1   = (const float*)d_in[2];
    const float* ew2   = (const float*)d_in[3];
    const float* eb2   = (const float*)d_in[4];
    const float* ew3   = (const float*)d_in[5];
    const float* eb3   = (const float*)d_in[6];
    const float* ew4   = (const float*)d_in[7];
    const float* eb4   = (const float*)d_in[8];
    const float* embed = (const float*)d_in[9];
    const float* dw1   = (const float*)d_in[10];
    const float* db1   = (const float*)d_in[11];
    const float* dtw1  = (const float*)d_in[12];
    const float* dtb1  = (const float*)d_in[13];
    const float* dtw2  = (const float*)d_in[14];
    const float* dtb2  = (const float*)d_in[15];
    const float* dtw3  = (const float*)d_in[16];
    const float* dtb3  = (const float*)d_in[17];
    (void)in_sizes; (void)n_in; (void)out_size; (void)ws_size;

    float* out = (float*)d_out;                       // [64*8192] dec, then [1] diff
    float* ws  = (float*)d_ws;

    size_t off = 0;
    float* bufA   = ws + off; off += (size_t)64 * 32 * 4096;   // h1, later g3
    float* bufB   = ws + off; off += (size_t)64 * 32 * 2048;   // h2, later g2
    float* bufC   = ws + off; off += (size_t)64 * 32 * 1024;   // h3, later g1
    float* zbuf   = ws + off; off += (size_t)64 * 1024 * 64;   // z rows [65536][64]
    float* qbuf   = ws + off; off += (size_t)64 * 64 * 1024;   // quant [b][d][t]
    float* embedT = ws + off; off += (size_t)2048 * 64;        // codebook, n-major
    float* e2     = ws + off; off += 2048;
    int*   idx    = (int*)(ws + off); off += 65536;

    float* diff = out + (size_t)64 * 8192;

    zero_scalar<<<1, 32, 0, stream>>>(diff);

    // codebook prep (independent of encoder; tiny)
    embed_transpose<<<512, 256, 0, stream>>>(embed, embedT);
    embed_norm     <<<8,   256, 0, stream>>>(embed, e2);

    // encoder
    enc_conv1  <<<32768, 256, 0, stream>>>(x, ew1, eb1, bufA);
    enc_conv_s2<<<16384, 256, 0, stream>>>(bufA, ew2, eb2, bufB, 4096, 2048);
    enc_conv_s2<<< 8192, 256, 0, stream>>>(bufB, ew3, eb3, bufC, 2048, 1024);
    enc_conv4  <<<16384, 256, 0, stream>>>(bufC, ew4, eb4, zbuf);

    // VQ (WMMA f32 16x16x4 matmul against codebook + argmin)
    vq_argmin<<<512,   256, 0, stream>>>(zbuf, embedT, e2, idx);
    vq_quant <<<16384, 256, 0, stream>>>(zbuf, embedT, idx, qbuf, diff);

    // decoder
    dec_conv1     <<< 8192, 256, 0, stream>>>(qbuf, dw1, db1, bufC);
    dec_convT     <<<16384, 256, 0, stream>>>(bufC, dtw1, dtb1, bufB, 1024);
    dec_convT     <<<32768, 256, 0, stream>>>(bufB, dtw2, dtb2, bufA, 2048);
    dec_convT_last<<< 2048, 256, 0, stream>>>(bufA, dtw3, dtb3, out, 4096);
}